// GCNStack_44951127720359
// MI455X (gfx1250) — compile-verified
//
#include <hip/hip_runtime.h>

typedef __attribute__((ext_vector_type(2))) float v2f;
typedef __attribute__((ext_vector_type(8))) float v8f;

#define NB    4
#define NN    50000
#define MM    (NB * NN)      // 200000 rows
#define CIN   64
#define HID   256
#define COUTN 64
#define NE    800000
#define STRIPS (MM / 16)     // 12500

// ---------------- Kernel 1: zero the x_mix accumulator (d_out) ----------------
__global__ void gcn_zero_kernel(float* __restrict__ y) {
  long i4 = ((long)blockIdx.x * blockDim.x + threadIdx.x) * 4;
  *(float4*)(y + i4) = make_float4(0.f, 0.f, 0.f, 0.f);
}

// ---------------- Kernel 2: COO scatter-add  y[b,r,:] += val * x[b,c,:] -------
__global__ void gcn_scatter_kernel(const float* __restrict__ x,
                                   const int* __restrict__ rows,
                                   const int* __restrict__ cols,
                                   const float* __restrict__ vals,
                                   float* __restrict__ y) {
  int t  = blockIdx.x * 256 + threadIdx.x;
  int e  = t >> 6;          // 64 channels per edge
  int ch = t & 63;
  if (e >= NE) return;
  int   r = rows[e];
  int   c = cols[e];
  float v = vals[e];
#pragma unroll
  for (int b = 0; b < NB; ++b) {
    float xv = x[(size_t)(b * NN + c) * CIN + ch];
    unsafeAtomicAdd(&y[(size_t)(b * NN + r) * CIN + ch], v * xv);
  }
}

// ---------------- Kernel 3: fused MLP, in-place over x_mix --------------------
// out[m,:] = relu(x_mix[m,:] @ W1 + b1) @ W2 + b2  using V_WMMA_F32_16X16X4_F32.
__global__ void __launch_bounds__(256, 1)
gcn_mlp_kernel(float* __restrict__ xo,          // [MM, 64] in: x_mix, out: result
               const float* __restrict__ W1,    // [64, 256]
               const float* __restrict__ b1,    // [256]
               const float* __restrict__ W2,    // [256, 64]
               const float* __restrict__ b2) {  // [64]
  __shared__ __align__(16) float sW1t[HID * CIN];    // [n][k]  = W1[k][n]
  __shared__ __align__(16) float sW2t[COUTN * HID];  // [o][k]  = W2[k][o]
  __shared__ __align__(16) float sB1[HID];
  __shared__ __align__(16) float sB2[COUTN];
  __shared__ __align__(16) float sStage[8][16 * 16]; // per-wave 16x16 bounce

  const int tid = threadIdx.x;
  for (int i = tid; i < CIN * HID; i += 256) {       // W1 [64,256] -> transposed
    int k = i >> 8, n = i & 255;
    sW1t[n * CIN + k] = W1[i];
  }
  for (int i = tid; i < HID * COUTN; i += 256) {     // W2 [256,64] -> transposed
    int k = i >> 6, o = i & 63;
    sW2t[o * HID + k] = W2[i];
  }
  if (tid < HID)   sB1[tid] = b1[tid];
  if (tid < COUTN) sB2[tid] = b2[tid];
  __syncthreads();

  const int wave  = tid >> 5;
  const int lane  = tid & 31;
  const int lhalf = lane >> 4;   // 0: lanes 0-15, 1: lanes 16-31
  const int lmod  = lane & 15;

  const int strip = blockIdx.x * 8 + wave;           // wave-uniform guard
  if (strip >= STRIPS) return;
  const int row0 = strip * 16;

  // A fragments for GEMM1: x_mix[row0:row0+16, 0:64], 16 K-steps of 4.
  // A(16x4 f32): lane holds M=lmod; v0 = K(4s+2*lhalf), v1 = K(4s+2*lhalf+1).
  v2f a1[16];
  {
    const float* base = xo + (size_t)(row0 + lmod) * CIN + 2 * lhalf;
#pragma unroll
    for (int s = 0; s < 16; ++s)
      a1[s] = *(const v2f*)(base + 4 * s);
  }

  // Output accumulators (16 rows x 64 cols = 4 N-tiles), seeded with b2.
  v8f oacc[4];
#pragma unroll
  for (int o = 0; o < 4; ++o) {
    float bv = sB2[o * 16 + lmod];
#pragma unroll
    for (int j = 0; j < 8; ++j) oacc[o][j] = bv;
  }

  float* st = &sStage[wave][0];

  for (int t = 0; t < 16; ++t) {
    // ---- GEMM1: h-tile = A_strip @ W1[:, 16t:16t+16] + b1 ----
    v8f h;
    float b1v = sB1[t * 16 + lmod];
#pragma unroll
    for (int j = 0; j < 8; ++j) h[j] = b1v;
#pragma unroll
    for (int s = 0; s < 16; ++s) {
      // B(4x16): lane holds N=lmod; v0/v1 = K rows (4s+2*lhalf, +1) -> contiguous in W1^T
      v2f bfrag = *(const v2f*)(&sW1t[(t * 16 + lmod) * CIN + 4 * s + 2 * lhalf]);
      h = __builtin_amdgcn_wmma_f32_16x16x4_f32(false, a1[s], false, bfrag,
                                                (short)0, h, false, false);
    }
#pragma unroll
    for (int j = 0; j < 8; ++j) h[j] = fmaxf(h[j], 0.f);

    // ---- C-layout -> row-major bounce through per-wave LDS tile ----
    // C: lanes 0-15 VGPR j -> (M=j, N=lmod); lanes 16-31 -> (M=8+j, N=lmod)
#pragma unroll
    for (int j = 0; j < 8; ++j)
      st[(lhalf * 8 + j) * 16 + lmod] = h[j];
    asm volatile("s_wait_dscnt 0" ::: "memory");

    // A fragments for GEMM2 from the staged tile (M=lmod, K runs 0..15)
    v2f a2[4];
#pragma unroll
    for (int kk = 0; kk < 4; ++kk)
      a2[kk] = *(const v2f*)(&st[lmod * 16 + 4 * kk + 2 * lhalf]);

    // ---- GEMM2: out += h_tile @ W2[16t:16t+16, :] ----
#pragma unroll
    for (int o = 0; o < 4; ++o) {
#pragma unroll
      for (int kk = 0; kk < 4; ++kk) {
        v2f b2frag = *(const v2f*)(&sW2t[(o * 16 + lmod) * HID + t * 16 + 4 * kk + 2 * lhalf]);
        oacc[o] = __builtin_amdgcn_wmma_f32_16x16x4_f32(false, a2[kk], false, b2frag,
                                                        (short)0, oacc[o], false, false);
      }
    }
  }

  // Epilogue: overwrite the strip in place (all A reads done long ago).
#pragma unroll
  for (int o = 0; o < 4; ++o) {
#pragma unroll
    for (int j = 0; j < 8; ++j)
      xo[(size_t)(row0 + lhalf * 8 + j) * COUTN + o * 16 + lmod] = oacc[o][j];
  }
}

extern "C" void kernel_launch(void* const* d_in, const int* in_sizes, int n_in,
                              void* d_out, int out_size, void* d_ws, size_t ws_size,
                              hipStream_t stream) {
  const float* x     = (const float*)d_in[0];
  const int*   A_idx = (const int*)d_in[1];   // [2, E]: rows then cols
  const float* A_val = (const float*)d_in[2];
  const float* W1    = (const float*)d_in[3];
  const float* b1    = (const float*)d_in[4];
  const float* W2    = (const float*)d_in[5];
  const float* b2    = (const float*)d_in[6];
  float* out = (float*)d_out;                 // [4, 50000, 64]; doubles as x_mix

  // 1) zero accumulator: 12,800,000 floats = 3,200,000 float4 = 12500 blocks
  gcn_zero_kernel<<<12500, 256, 0, stream>>>(out);

  // 2) scatter-add sparse mix into d_out: 800k edges * 64 ch / 256 thr
  gcn_scatter_kernel<<<(NE * 64) / 256, 256, 0, stream>>>(
      x, A_idx, A_idx + NE, A_val, out);

  // 3) fused WMMA MLP in place: 12500 strips, 8 strips (waves) per block
  gcn_mlp_kernel<<<(STRIPS + 7) / 8, 256, 0, stream>>>(out, W1, b1, W2, b2);
}